// YOLOloss_22033182229033
// MI455X (gfx1250) — compile-verified
//
#include <hip/hip_runtime.h>

// ---------------- problem constants ----------------
#define SS          49            // S*S cells per sample
#define PRED_F      30            // floats per cell in predictions (C + 2*5)
#define TGT_F       25            // floats per cell in targets (C + 5)
#define NCLS        20
#define L_COORD     5.0f
#define L_NOOBJ     0.5f
#define EPS_F       1e-6f

// ---------------- tiling ----------------
#define THREADS     256
#define TILE        256                       // cells per tile (== THREADS)
#define PRED_DW     (TILE * PRED_F)           // 7680 dwords
#define TGT_DW      (TILE * TGT_F)            // 6400 dwords
#define TILE_DW     (PRED_DW + TGT_DW)        // 14080 dwords
#define PRED_CHUNKS (PRED_DW / 4)             // 1920 x b128
#define CHUNKS      (TILE_DW / 4)             // 3520 x b128
#define KPT         ((CHUNKS + THREADS - 1) / THREADS)   // 14 asyncs / thread / tile
#define GRID_BLKS   1024

// ---------------- CDNA5 async global->LDS path ----------------
#if defined(__gfx1250__) && \
    __has_builtin(__builtin_amdgcn_global_load_async_to_lds_b128) && \
    __has_builtin(__builtin_amdgcn_s_wait_asynccnt)
#define USE_ASYNC 1
#else
#define USE_ASYNC 0
#endif

#if USE_ASYNC
typedef int v4i __attribute__((vector_size(16)));
typedef __attribute__((address_space(1))) v4i* glb_v4i_p;
typedef __attribute__((address_space(3))) v4i* lds_v4i_p;
__device__ __forceinline__ void cp16_async(const float* g, float* l) {
    // global pointers are identity-mapped in generic space -> inttoptr to as(1);
    // LDS generic pointer -> as(3) via addrspacecast.
    __builtin_amdgcn_global_load_async_to_lds_b128(
        (glb_v4i_p)(unsigned long long)(const void*)g,
        (lds_v4i_p)l, 0, 0);
}
#define WAIT_ASYNC(n) __builtin_amdgcn_s_wait_asynccnt(n)
#else
__device__ __forceinline__ void cp16_async(const float* g, float* l) {
    *(float4*)l = *(const float4*)g;   // synchronous fallback
}
#define WAIT_ASYNC(n) ((void)0)
#endif

// ---------------- per-cell math ----------------
__device__ __forceinline__ float iou4(const float* a, const float* b) {
    float ax1 = a[0] - a[2] * 0.5f, ay1 = a[1] - a[3] * 0.5f;
    float ax2 = a[0] + a[2] * 0.5f, ay2 = a[1] + a[3] * 0.5f;
    float bx1 = b[0] - b[2] * 0.5f, by1 = b[1] - b[3] * 0.5f;
    float bx2 = b[0] + b[2] * 0.5f, by2 = b[1] + b[3] * 0.5f;
    float iw = fmaxf(fminf(ax2, bx2) - fmaxf(ax1, bx1), 0.0f);
    float ih = fmaxf(fminf(ay2, by2) - fmaxf(ay1, by1), 0.0f);
    float inter  = iw * ih;
    float area_a = fabsf((ax2 - ax1) * (ay2 - ay1));
    float area_b = fabsf((bx2 - bx1) * (by2 - by1));
    return inter / (area_a + area_b - inter + EPS_F);
}

__device__ __forceinline__ float cell_loss(const float* __restrict__ p,
                                           const float* __restrict__ q) {
    float b1[5], b2[5], tb[5];
#pragma unroll
    for (int i = 0; i < 5; ++i) {
        b1[i] = p[NCLS + i];
        b2[i] = p[NCLS + 5 + i];
        tb[i] = q[NCLS + i];
    }
    float obj = (tb[0] == 1.0f) ? 1.0f : 0.0f;
    float i1 = iou4(b1, tb);
    float i2 = iou4(b2, tb);
    bool  r1 = i1 > i2;
    float r[5];
#pragma unroll
    for (int i = 0; i < 5; ++i) r[i] = r1 ? b1[i] : b2[i];
    float dx = r[0] - tb[0], dy = r[1] - tb[1];
    float xy = dx * dx + dy * dy;
    float sw = __fsqrt_rn(r[2]) - __fsqrt_rn(tb[2]);
    float sh = __fsqrt_rn(r[3]) - __fsqrt_rn(tb[3]);
    float wh = sw * sw + sh * sh;
    float coord = L_COORD * (xy + wh);
    float dc = r[4] - tb[4];
    float conf = dc * dc;
    float cls = 0.0f;
#pragma unroll
    for (int i = 0; i < NCLS; ++i) {
        float d = p[i] - q[i];
        cls = fmaf(d, d, cls);
    }
    float noobj = L_NOOBJ * (b1[4] * b1[4] + b2[4] * b2[4]);
    return obj * (coord + conf + cls) + (1.0f - obj) * noobj;
}

// ---------------- staging: one tile (256 cells) -> LDS buffer ----------------
__device__ __forceinline__ void stage_tile(const float* __restrict__ pred,
                                           const float* __restrict__ tgt,
                                           float* __restrict__ smem,
                                           int bufSel, long tile,
                                           long predDwTot, long tgtDwTot,
                                           int tid) {
    const float* gp = pred + tile * PRED_DW;
    const float* gq = tgt  + tile * TGT_DW;
    float* lb    = smem + (size_t)bufSel * TILE_DW;
    float* ldump = smem + 2 * (size_t)TILE_DW;   // scrap slot for dummy issues
#pragma unroll
    for (int k = 0; k < KPT; ++k) {
        int j = tid + k * THREADS;
        const float* g = pred;       // default: harmless re-read of array start
        float*       l = ldump;      // default: scrap LDS slot
        if (j < PRED_CHUNKS) {
            long dw = tile * PRED_DW + (long)j * 4;
            if (dw + 4 <= predDwTot) { g = gp + (long)j * 4; l = lb + j * 4; }
        } else if (j < CHUNKS) {
            int  jj = j - PRED_CHUNKS;
            long dw = tile * TGT_DW + (long)jj * 4;
            if (dw + 4 <= tgtDwTot) { g = gq + (long)jj * 4; l = lb + PRED_DW + jj * 4; }
        }
        cp16_async(g, l);            // every thread issues exactly KPT b128 asyncs
    }
}

// ---------------- kernel 1: per-block partial sums ----------------
__global__ __launch_bounds__(THREADS)
void yolo_partial_kernel(const float* __restrict__ pred,
                         const float* __restrict__ tgt,
                         float* __restrict__ partial,
                         int numTiles, long totalCells) {
    extern __shared__ float smem[];              // 2*TILE_DW + 4 floats
    __shared__ float wsum[THREADS / 32];
    const int tid = threadIdx.x;
    const long predDwTot = totalCells * PRED_F;
    const long tgtDwTot  = totalCells * TGT_F;

    float acc = 0.0f;
    long t = blockIdx.x;
    int cur = 0;
    if (t < numTiles)
        stage_tile(pred, tgt, smem, 0, t, predDwTot, tgtDwTot, tid);

    while (t < numTiles) {
        long nxt = t + gridDim.x;
        if (nxt < numTiles) {
            stage_tile(pred, tgt, smem, cur ^ 1, nxt, predDwTot, tgtDwTot, tid);
            WAIT_ASYNC(KPT);         // tile t's KPT asyncs complete; nxt's in flight
        } else {
            WAIT_ASYNC(0);
        }
        __syncthreads();             // all waves' portions of tile t visible

        long cell = t * TILE + tid;
        if (cell < totalCells) {
            const float* p = smem + (size_t)cur * TILE_DW + tid * PRED_F;
            const float* q = smem + (size_t)cur * TILE_DW + PRED_DW + tid * TGT_F;
            acc += cell_loss(p, q);
        }
        __syncthreads();             // done reading before buffer is reused
        cur ^= 1;
        t = nxt;
    }

    // wave32 reduction, then deterministic per-block sum
#pragma unroll
    for (int off = 16; off > 0; off >>= 1)
        acc += __shfl_down(acc, off, 32);
    if ((tid & 31) == 0) wsum[tid >> 5] = acc;
    __syncthreads();
    if (tid == 0) {
        float s = 0.0f;
#pragma unroll
        for (int w = 0; w < THREADS / 32; ++w) s += wsum[w];
        partial[blockIdx.x] = s;
    }
}

// ---------------- kernel 2: deterministic final reduction ----------------
__global__ __launch_bounds__(256)
void yolo_final_kernel(const float* __restrict__ partial, int n,
                       float* __restrict__ out, float invN) {
    __shared__ float s[256];
    float a = 0.0f;
    for (int i = threadIdx.x; i < n; i += 256) a += partial[i];
    s[threadIdx.x] = a;
    __syncthreads();
    for (int k = 128; k > 0; k >>= 1) {
        if (threadIdx.x < k) s[threadIdx.x] += s[threadIdx.x + k];
        __syncthreads();
    }
    if (threadIdx.x == 0) out[0] = s[0] * invN;
}

// ---------------- launch ----------------
extern "C" void kernel_launch(void* const* d_in, const int* in_sizes, int n_in,
                              void* d_out, int out_size, void* d_ws, size_t ws_size,
                              hipStream_t stream) {
    const float* pred = (const float*)d_in[0];
    const float* tgt  = (const float*)d_in[1];
    float* out = (float*)d_out;
    float* partial = (float*)d_ws;

    const long N          = (long)in_sizes[0] / (SS * PRED_F);   // 16384
    const long totalCells = N * SS;                              // 802816
    const int  numTiles   = (int)((totalCells + TILE - 1) / TILE);
    const int  grid       = numTiles < GRID_BLKS ? numTiles : GRID_BLKS;
    const size_t shmem    = (2 * (size_t)TILE_DW + 4) * sizeof(float); // 112656 B

    yolo_partial_kernel<<<grid, THREADS, shmem, stream>>>(
        pred, tgt, partial, numTiles, totalCells);
    yolo_final_kernel<<<1, 256, 0, stream>>>(
        partial, grid, out, 1.0f / (float)N);
}